// HungarianLoss_23081154248911
// MI455X (gfx1250) — compile-verified
//
#include <hip/hip_runtime.h>
#include <hip/hip_bf16.h>
#include <math.h>

#define BATCH 128
#define NQ    200
#define NC    92
#define NP1   201   // columns incl. sentinel
#define TILES 13    // ceil(200/16)
#define SLOTS 7     // ceil(201/32)

typedef __attribute__((ext_vector_type(2))) float v2f;
typedef __attribute__((ext_vector_type(4))) float v4f;
typedef __attribute__((ext_vector_type(8))) float v8f;

// ---------------------------------------------------------------------------
// Kernel 1: per-(b,i) logsumexp over the 92 classes (max-subtracted, matching
// jax.nn.log_softmax exactly in formula).
// ---------------------------------------------------------------------------
__global__ __launch_bounds__(256) void lse_kernel(const float* __restrict__ pred_cat,
                                                  float* __restrict__ lse) {
    int t = blockIdx.x * 256 + threadIdx.x;
    if (t >= BATCH * NQ) return;
    const float* row = pred_cat + (size_t)t * NC;
    float m = row[0];
    #pragma unroll 4
    for (int c = 1; c < NC; ++c) m = fmaxf(m, row[c]);
    float s = 0.f;
    #pragma unroll 4
    for (int c = 0; c < NC; ++c) s += expf(row[c] - m);
    lse[t] = m + logf(s);
}

// ---------------------------------------------------------------------------
// Kernel 2: cost matrix. CE gather realized as logits @ onehot(tar_cat)^T via
// V_WMMA_F32_16X16X4_F32 (one-hot B => bit-exact f32 gather). One wave per
// 16x16 output tile. Epilogue fuses lse - gathered + SmoothL1*mask.
//
// Layouts per CDNA5 ISA 7.12.2:
//   A 16x4 f32 : lane l: M = l&15 ; VGPR0/1 = K = 2*(l>>4) + {0,1}
//   B 4x16 f32 : lane l: N = l&15 ; VGPR0/1 = K = 2*(l>>4) + {0,1}
//   C/D 16x16  : VGPR r: M = r + 8*(l>>4), N = l&15
//
// Padded rows (m >= 200) read row 0 instead (clamped pointer): the products
// land in D rows that are never stored, so no zero-fill / predication needed
// -> plain global_load_b64 per K-pair, free to pipeline past the WMMAs.
// ---------------------------------------------------------------------------
__global__ __launch_bounds__(32) void cost_wmma_kernel(const float* __restrict__ pred_cat,
                                                       const float* __restrict__ pred_bbox,
                                                       const int*   __restrict__ tar_cat,
                                                       const float* __restrict__ tar_bbox,
                                                       const float* __restrict__ lse,
                                                       float*       __restrict__ cost) {
    int blk  = blockIdx.x;
    int b    = blk / (TILES * TILES);
    int t    = blk % (TILES * TILES);
    int mi   = t / TILES;           // row tile
    int nj   = t % TILES;           // col tile
    int lane = threadIdx.x;
    int half = lane >> 4;           // 0: K pair {0,1}, 1: K pair {2,3}
    int l15  = lane & 15;

    int m = mi * 16 + l15;          // A row this lane feeds (0..207)
    int n = nj * 16 + l15;          // B col this lane feeds (0..207)
    bool nvalid = (n < NQ);

    int tc = nvalid ? tar_cat[b * NQ + n] : -1;        // one-hot class, -1 = pad
    const float* arow = pred_cat + ((size_t)b * NQ + (m < NQ ? m : 0)) * NC
                      + 2 * half;                      // this lane's K-pair base

    v8f acc = {};
    #pragma unroll
    for (int k0 = 0; k0 < NC; k0 += 4) {               // 92 = 23 * 4, no K pad
        int kk = k0 + 2 * half;
        v2f a = *(const v2f*)(arow + k0);              // aligned 8B -> b64 load
        v2f bb;
        bb.x = (tc == kk)     ? 1.f : 0.f;
        bb.y = (tc == kk + 1) ? 1.f : 0.f;
        acc = __builtin_amdgcn_wmma_f32_16x16x4_f32(
            /*neg_a=*/false, a, /*neg_b=*/false, bb,
            /*c_mod=*/(short)0, acc, /*reuse_a=*/false, /*reuse_b=*/false);
    }

    if (!nvalid) return;
    // Per-lane fixed target column n: bbox (b128 load) + mask
    v4f tb = *(const v4f*)(tar_bbox + ((size_t)b * NQ + n) * 4);
    float maskf = (tc != 0) ? 1.f : 0.f;

    #pragma unroll
    for (int r = 0; r < 8; ++r) {
        int mrow = mi * 16 + half * 8 + r;             // D-matrix row for acc[r]
        if (mrow < NQ) {
            float g = acc[r];                          // gathered logit (exact)
            float l = lse[b * NQ + mrow];
            v4f pb = *(const v4f*)(pred_bbox + ((size_t)b * NQ + mrow) * 4);
            float d0 = pb.x - tb.x, d1 = pb.y - tb.y, d2 = pb.z - tb.z, d3 = pb.w - tb.w;
            float a0 = fabsf(d0), a1 = fabsf(d1), a2 = fabsf(d2), a3 = fabsf(d3);
            float sl1 = (a0 < 1.f ? 0.5f * d0 * d0 : a0 - 0.5f)
                      + (a1 < 1.f ? 0.5f * d1 * d1 : a1 - 0.5f)
                      + (a2 < 1.f ? 0.5f * d2 * d2 : a2 - 0.5f)
                      + (a3 < 1.f ? 0.5f * d3 * d3 : a3 - 0.5f);
            cost[((size_t)b * NQ + mrow) * NQ + n] = (l - g) + sl1 * maskf;
        }
    }
}

// ---------------------------------------------------------------------------
// Kernel 3: exact JV Hungarian (e-maxx form, replicating the reference
// semantics incl. first-index argmin tie-break), one wave32 per batch.
// All state in LDS; columns distributed 7-per-lane; argmin via shfl_xor.
// Single-wave workgroup => __syncthreads is just waitcnt + S_NOP.
// Finishes by scattering the matched cost per row into d_out.
// ---------------------------------------------------------------------------
__global__ __launch_bounds__(32) void hungarian_kernel(const float* __restrict__ cost,
                                                       float* __restrict__ out) {
    __shared__ float u[NP1], v[NP1], minv[NP1];
    __shared__ int   p[NP1], way[NP1], used[NP1];

    int b    = blockIdx.x;
    int lane = threadIdx.x;
    const float* cb = cost + (size_t)b * NQ * NQ;

    for (int s = 0; s < SLOTS; ++s) {
        int j = lane + 32 * s;
        if (j < NP1) { u[j] = 0.f; v[j] = 0.f; p[j] = 0; }
    }
    __syncthreads();

    for (int i = 1; i <= NQ; ++i) {
        for (int s = 0; s < SLOTS; ++s) {
            int j = lane + 32 * s;
            if (j < NP1) { minv[j] = 1e9f; way[j] = 0; used[j] = 0; }
        }
        if (lane == 0) p[0] = i;
        __syncthreads();

        int j0 = 0;
        while (true) {
            int i0 = p[j0];                 // uniform LDS broadcast read
            if (i0 == 0) break;
            if (lane == 0) used[j0] = 1;
            __syncthreads();

            float ui0 = u[i0];
            const float* crow = cb + (size_t)(i0 - 1) * NQ;

            float bestv = 3.0e38f;
            int   bestj = 0;
            for (int s = 0; s < SLOTS; ++s) {
                int j = lane + 32 * s;
                if (j >= 1 && j < NP1) {
                    float mj = minv[j];
                    if (!used[j]) {
                        float cur = crow[j - 1] - ui0 - v[j];
                        if (cur < mj) { mj = cur; minv[j] = mj; way[j] = j0; }
                    }
                    float masked = used[j] ? 1e9f : mj;
                    if (masked < bestv || (masked == bestv && j < bestj)) {
                        bestv = masked; bestj = j;
                    }
                }
            }
            // wave-wide argmin, reference tie-break (smallest column index)
            for (int off = 16; off >= 1; off >>= 1) {
                float ov = __shfl_xor(bestv, off, 32);
                int   oj = __shfl_xor(bestj, off, 32);
                if (ov < bestv || (ov == bestv && oj < bestj)) { bestv = ov; bestj = oj; }
            }
            float delta = bestv;
            int   j1    = bestj;

            __syncthreads();
            for (int s = 0; s < SLOTS; ++s) {
                int j = lane + 32 * s;
                if (j < NP1) {
                    if (used[j]) {
                        v[j] -= delta;
                        u[p[j]] += delta;   // rows on alternating tree are distinct
                    } else {
                        minv[j] -= delta;
                    }
                }
            }
            __syncthreads();
            j0 = j1;
        }

        if (lane == 0) {                    // augment along alternating path
            int jj = j0;
            while (jj != 0) { int jn = way[jj]; p[jj] = p[jn]; jj = jn; }
        }
        __syncthreads();
    }

    // out[b, p[j]-1] = cost[b, p[j]-1, j-1]  (p is a permutation: covers all rows)
    for (int s = 0; s < SLOTS; ++s) {
        int j = lane + 32 * s;
        if (j >= 1 && j < NP1) {
            int r = p[j] - 1;
            out[(size_t)b * NQ + r] = cb[(size_t)r * NQ + (j - 1)];
        }
    }
}

// ---------------------------------------------------------------------------
extern "C" void kernel_launch(void* const* d_in, const int* in_sizes, int n_in,
                              void* d_out, int out_size, void* d_ws, size_t ws_size,
                              hipStream_t stream) {
    const float* pred_cat  = (const float*)d_in[0];   // [128,200,92]
    const float* pred_bbox = (const float*)d_in[1];   // [128,200,4]
    const int*   tar_cat   = (const int*)  d_in[2];   // [128,200]
    const float* tar_bbox  = (const float*)d_in[3];   // [128,200,4]
    float*       out       = (float*)d_out;           // [128,200]

    float* cost = (float*)d_ws;                       // 128*200*200 f32 = 20.48 MB
    float* lse  = cost + (size_t)BATCH * NQ * NQ;     // 128*200 f32

    // 1) logsumexp per (b, i)
    {
        int total  = BATCH * NQ;
        int blocks = (total + 255) / 256;
        lse_kernel<<<blocks, 256, 0, stream>>>(pred_cat, lse);
    }
    // 2) cost matrix: one wave per 16x16 tile, 13x13 tiles per batch
    {
        int blocks = BATCH * TILES * TILES;
        cost_wmma_kernel<<<blocks, 32, 0, stream>>>(pred_cat, pred_bbox, tar_cat,
                                                    tar_bbox, lse, cost);
    }
    // 3) Hungarian + matched-cost gather: one wave per batch
    {
        hungarian_kernel<<<BATCH, 32, 0, stream>>>(cost, out);
    }
}